// EdgeToNodeElementWis_5420248727809
// MI455X (gfx1250) — compile-verified
//
#include <hip/hip_runtime.h>
#include <hip/hip_bf16.h>

typedef __attribute__((ext_vector_type(2))) float v2f;
typedef __attribute__((ext_vector_type(8))) float v8f;

#define N_BATCH 4096
#define P_OUT   64
#define D_SP    90
#define DD2     (D_SP * D_SP)   // 8100

// x: (N,1,D,D) f32, w: (P,1,D,D) f32, out: (N,P,D,1) f32
// out[n,p,i] = sum_j x[n,0,i,j] * w[p,0,i,j]
// One block = 4 waves; each wave does a 16(n) x 16(p) tile for one slice i
// via 23 chained v_wmma_f32_16x16x4_f32 (K = 90 = 22*4 + 2, zero-padded tail).
// K loop is fully unrolled: all loads become global_load_b64 with literal
// immediate offsets off two base pointers -> zero per-iteration address math.
__global__ __launch_bounds__(128)
void slice_gemm_wmma_f32(const float* __restrict__ x,
                         const float* __restrict__ w,
                         float* __restrict__ out) {
  const int i    = blockIdx.x;          // spatial row 0..89 (one GEMM per i)
  const int n0   = blockIdx.y << 4;     // 16-row n tile
  const int lane = threadIdx.x & 31;
  const int wave = threadIdx.x >> 5;    // 0..3 -> p tile
  const int p0   = wave << 4;

  const int rc = lane & 15;             // A row / B col within the tile
  const int kh = lane >> 4;             // half-wave selects K pair {0,1} vs {2,3}

  // A fragment source: x[n0+rc, i, 4*k + 2*kh + {0,1}]
  const float* ap = x + (size_t)(n0 + rc) * DD2 + (size_t)i * D_SP + 2 * kh;
  // B fragment source: w[p0+rc, i, 4*k + 2*kh + {0,1}]  (B = W_i^T, N-dim = p)
  const float* bp = w + (size_t)(p0 + rc) * DD2 + (size_t)i * D_SP + 2 * kh;

  v8f c = {};

  // Main K chain: 22 steps cover j = 0..87 (all lanes in-bounds).
#pragma unroll
  for (int k = 0; k < 22; ++k) {
    v2f a = *(const v2f*)(ap + 4 * k);
    v2f b = *(const v2f*)(bp + 4 * k);
    c = __builtin_amdgcn_wmma_f32_16x16x4_f32(
        /*neg_a=*/false, a, /*neg_b=*/false, b,
        /*c_mod=*/(short)0, c, /*reuse_a=*/false, /*reuse_b=*/false);
  }

  // Tail: j = 88 + 2*kh (+0/1). kh==1 lanes would read j=90,91 (OOB):
  // clamp their address back to j=88,89 and zero the values so the WMMA
  // executes with EXEC all-1s and the padded K columns contribute 0.
  {
    v2f a = *(const v2f*)(ap + 88 - 2 * kh);
    v2f b = *(const v2f*)(bp + 88 - 2 * kh);
    if (kh) {
      a = (v2f){0.f, 0.f};
      b = (v2f){0.f, 0.f};
    }
    c = __builtin_amdgcn_wmma_f32_16x16x4_f32(false, a, false, b,
                                              (short)0, c, false, false);
  }

  // C/D layout: VGPR r -> lanes 0-15: (M=r, N=lane); lanes 16-31: (M=8+r, N=lane-16).
  // out index: ((n * P_OUT + p) * D_SP + i)
  const int pcol  = p0 + rc;
  const int mbase = n0 + (kh << 3);
  float* o = out + ((size_t)mbase * P_OUT + pcol) * D_SP + i;
#pragma unroll
  for (int r = 0; r < 8; ++r) {
    o[(size_t)r * P_OUT * D_SP] = c[r];
  }
}

extern "C" void kernel_launch(void* const* d_in, const int* in_sizes, int n_in,
                              void* d_out, int out_size, void* d_ws, size_t ws_size,
                              hipStream_t stream) {
  const float* x = (const float*)d_in[0];   // (4096,1,90,90) f32
  const float* w = (const float*)d_in[1];   // (64,1,90,90)  f32
  float* out = (float*)d_out;               // (4096,64,90,1) f32

  dim3 grid(D_SP, N_BATCH / 16);            // (90, 256)
  dim3 block(128);                          // 4 wave32s
  slice_gemm_wmma_f32<<<grid, block, 0, stream>>>(x, w, out);
}